// SoftPerVertexSHShader_21062519619816
// MI455X (gfx1250) — compile-verified
//
#include <hip/hip_runtime.h>
#include <math.h>

typedef __attribute__((ext_vector_type(2))) float v2f;
typedef __attribute__((ext_vector_type(8))) float v8f;

// ---------------------------------------------------------------------------
// Kernel 1: per-vertex SH shading via V_WMMA_F32_16X16X4_F32.
//   shading = Y(V x 9) @ sh(9 x 3), K padded 9->12 (3 chained WMMAs),
//   N padded 3->16;  shaded_color = verts_colors * shading.
// One wave per 16-vertex tile. All fragment element indices are compile-time
// constants; the lane-half (hi) dependence is a single select per element.
// EXEC stays all-ones across the WMMAs (tail tile clamps loads, guards stores).
// ---------------------------------------------------------------------------
__global__ void sh_shade_wmma_kernel(const float* __restrict__ vn,
                                     const float* __restrict__ vc,
                                     const float* __restrict__ sh,
                                     float* __restrict__ shaded,
                                     int V)
{
  const int gwave = (int)((blockIdx.x * blockDim.x + threadIdx.x) >> 5);
  const int lane  = (int)(threadIdx.x & 31);
  const int base  = gwave * 16;
  if (base >= V) return;                 // uniform per wave

  const int  col = lane & 15;            // M row for A, N column for B/D
  const bool hi  = lane >= 16;

  int vid = base + col;
  if (vid >= V) vid = V - 1;             // clamp: keep EXEC all-ones

  float nx = vn[vid * 3 + 0];
  float ny = vn[vid * 3 + 1];
  float nz = vn[vid * 3 + 2];
  float nrm = fmaxf(sqrtf(nx * nx + ny * ny + nz * nz), 1e-6f);
  const float inv = 1.0f / nrm;
  nx *= inv; ny *= inv; nz *= inv;

  const float A0C0 = 0.886226925452758f;   // pi / sqrt(4 pi)
  const float A1C1 = 1.772453850905516f;   // sqrt(pi)
  const float A2C2 = 2.427032383628256f;   // (pi/sqrt(2)) * 3 sqrt(5)/sqrt(12 pi)

  float Y[12];
  Y[0]  =  A0C0;
  Y[1]  = -A1C1 * ny;
  Y[2]  =  A1C1 * nz;
  Y[3]  = -A1C1 * nx;
  Y[4]  =  A2C2 * nx * ny;
  Y[5]  = -A2C2 * ny * nz;
  Y[6]  =  A2C2 * (0.5f / 1.7320508075688772f) * (3.0f * nz * nz - 1.0f);
  Y[7]  = -A2C2 * nx * nz;
  Y[8]  =  A2C2 * 0.5f * (nx * nx - ny * ny);
  Y[9]  = 0.0f; Y[10] = 0.0f; Y[11] = 0.0f;

  // Per-lane B-matrix column: B[k] = sh[k][col] (+ambient on band 0), zero pad.
  const bool cok = col < 3;
  const int  cc  = cok ? col : 0;        // clamp for unconditional loads
  float B[12];
#pragma unroll
  for (int k = 0; k < 9; ++k) {
    float v = sh[k * 3 + cc];
    if (k == 0) v += 0.8f;               // AMBIENT
    B[k] = cok ? v : 0.0f;
  }
  B[9] = 0.0f; B[10] = 0.0f; B[11] = 0.0f;

  v8f d = {0.f, 0.f, 0.f, 0.f, 0.f, 0.f, 0.f, 0.f};
#pragma unroll
  for (int c = 0; c < 3; ++c) {
    // A 16x4 layout: lanes 0-15 carry K=4c+0/4c+1, lanes 16-31 carry 4c+2/4c+3.
    // B 4x16 mirrors it. Constant indices -> one v_cndmask per element.
    v2f a, b;
    a.x = hi ? Y[4 * c + 2] : Y[4 * c + 0];
    a.y = hi ? Y[4 * c + 3] : Y[4 * c + 1];
    b.x = hi ? B[4 * c + 2] : B[4 * c + 0];
    b.y = hi ? B[4 * c + 3] : B[4 * c + 1];
    d = __builtin_amdgcn_wmma_f32_16x16x4_f32(false, a, false, b,
                                              (short)0, d, false, false);
  }

  // D layout: VGPR j -> (M = j + (hi?8:0), N = col). Only RGB columns are real.
  if (cok) {
    const int mbase = base + (hi ? 8 : 0);
#pragma unroll
    for (int j = 0; j < 8; ++j) {
      const int m = mbase + j;
      if (m < V) shaded[m * 3 + col] = vc[m * 3 + col] * d[j];
    }
  }
}

// ---------------------------------------------------------------------------
// Kernel 2: gather + barycentric interp + softmax RGB blend.
// One thread per pixel; the K=8 fragment data per pixel is contiguous, so we
// pull it with int4/float4 loads (bandwidth-bound: ~117 MB streamed; the
// faces/shaded-color gather tables are 3.6 MB total -> resident in L2).
// ---------------------------------------------------------------------------
__global__ void soft_blend_kernel(const float* __restrict__ shaded,
                                  const int*   __restrict__ faces,
                                  const float* __restrict__ bary,
                                  const float* __restrict__ zbuf,
                                  const float* __restrict__ dists,
                                  const int*   __restrict__ p2f,
                                  float* __restrict__ out,
                                  int NP)
{
  const int p = (int)(blockIdx.x * blockDim.x + threadIdx.x);
  if (p >= NP) return;

  const float SIGMA_INV = 1.0f / 1e-4f;
  const float GAMMA_INV = 1.0f / 1e-4f;
  const float EPS   = 1e-10f;
  const float ZFAR  = 100.0f;
  const float ZNEAR = 1.0f;
  const float ZSCALE = 1.0f / (ZFAR - ZNEAR);

  const int4   f0 = *(const int4*  )(p2f   + (size_t)p * 8);
  const int4   f1 = *(const int4*  )(p2f   + (size_t)p * 8 + 4);
  const float4 z0 = *(const float4*)(zbuf  + (size_t)p * 8);
  const float4 z1 = *(const float4*)(zbuf  + (size_t)p * 8 + 4);
  const float4 q0 = *(const float4*)(dists + (size_t)p * 8);
  const float4 q1 = *(const float4*)(dists + (size_t)p * 8 + 4);

  float bw[24];
#pragma unroll
  for (int i = 0; i < 6; ++i)
    *(float4*)(bw + 4 * i) = *(const float4*)(bary + (size_t)p * 24 + 4 * i);

  const int   fk[8] = {f0.x, f0.y, f0.z, f0.w, f1.x, f1.y, f1.z, f1.w};
  const float zk[8] = {z0.x, z0.y, z0.z, z0.w, z1.x, z1.y, z1.z, z1.w};
  const float dk[8] = {q0.x, q0.y, q0.z, q0.w, q1.x, q1.y, q1.z, q1.w};

  float cr[8], cg[8], cb[8], zi[8], pr[8];
  float alpha = 1.0f;
  float zmax  = EPS;

#pragma unroll
  for (int k = 0; k < 8; ++k) {
    const int  f = fk[k];
    const bool m = (f >= 0);
    float r = 0.0f, g = 0.0f, b = 0.0f;
    if (m) {
      const int i0 = faces[f * 3 + 0];
      const int i1 = faces[f * 3 + 1];
      const int i2 = faces[f * 3 + 2];
      const float w0 = bw[k * 3 + 0];
      const float w1 = bw[k * 3 + 1];
      const float w2 = bw[k * 3 + 2];
      r = w0 * shaded[i0 * 3 + 0] + w1 * shaded[i1 * 3 + 0] + w2 * shaded[i2 * 3 + 0];
      g = w0 * shaded[i0 * 3 + 1] + w1 * shaded[i1 * 3 + 1] + w2 * shaded[i2 * 3 + 1];
      b = w0 * shaded[i0 * 3 + 2] + w1 * shaded[i1 * 3 + 2] + w2 * shaded[i2 * 3 + 2];
    }
    cr[k] = r; cg[k] = g; cb[k] = b;

    // prob = sigmoid(-d/sigma) * mask ; expf underflow -> 0 matches jnp.exp
    const float prob = m ? 1.0f / (1.0f + expf(dk[k] * SIGMA_INV)) : 0.0f;
    pr[k] = prob;
    alpha *= (1.0f - prob);

    const float z = m ? (ZFAR - zk[k]) * ZSCALE : 0.0f;
    zi[k] = z;
    zmax  = fmaxf(zmax, z);
  }

  const float delta = expf((EPS - zmax) * GAMMA_INV);   // background weight
  float denom = delta;
  float wr = delta, wg = delta, wb = delta;             // BG = (1,1,1)
#pragma unroll
  for (int k = 0; k < 8; ++k) {
    const float w = pr[k] * expf((zi[k] - zmax) * GAMMA_INV);
    denom += w;
    wr += w * cr[k];
    wg += w * cg[k];
    wb += w * cb[k];
  }
  const float dinv = 1.0f / denom;
  *(float4*)(out + (size_t)p * 4) =
      make_float4(wr * dinv, wg * dinv, wb * dinv, 1.0f - alpha);
}

// ---------------------------------------------------------------------------

extern "C" void kernel_launch(void* const* d_in, const int* in_sizes, int n_in,
                              void* d_out, int out_size, void* d_ws, size_t ws_size,
                              hipStream_t stream)
{
  const float* vn    = (const float*)d_in[0];   // verts_normals (V,3)
  const float* vc    = (const float*)d_in[1];   // verts_colors  (V,3)
  const float* sh    = (const float*)d_in[2];   // sh_coeffs     (1,9,3)
  const float* bary  = (const float*)d_in[3];   // (N,H,W,K,3)
  const float* zbuf  = (const float*)d_in[4];   // (N,H,W,K)
  const float* dists = (const float*)d_in[5];   // (N,H,W,K)
  const int*   faces = (const int*)  d_in[6];   // (F,3)
  const int*   p2f   = (const int*)  d_in[7];   // (N,H,W,K)
  float*       out   = (float*)d_out;           // (N,H,W,4)
  float*       shaded = (float*)d_ws;           // scratch: (V,3) shaded colors

  const int V  = in_sizes[0] / 3;
  const int NP = in_sizes[7] / 8;               // K = 8

  // One wave per 16-vertex WMMA tile; 8 waves (256 threads) per block.
  const int waves   = (V + 15) / 16;
  const int blocks1 = (waves + 7) / 8;
  sh_shade_wmma_kernel<<<blocks1, 256, 0, stream>>>(vn, vc, sh, shaded, V);

  const int blocks2 = (NP + 255) / 256;
  soft_blend_kernel<<<blocks2, 256, 0, stream>>>(shaded, faces, bary, zbuf,
                                                 dists, p2f, out, NP);
}